// ArmInt_19911468384433
// MI455X (gfx1250) — compile-verified
//
#include <hip/hip_runtime.h>
#include <math.h>

// CDNA5 / gfx1250: wave32, WMMA F32 16x16x4.
typedef __attribute__((ext_vector_type(2))) float v2f;
typedef __attribute__((ext_vector_type(8))) float v8f;

#define FPFM 256.0f
#define LDSTRIDE 36   // padded row stride in floats: 36*4=144 bytes (16B aligned), odd*4 bank spread

__device__ __forceinline__ float round_div_fpfm(float xx) {
    // xx + sign(xx)*128, then trunc(xx/256).  copysign differs from sign() only at xx==0,
    // where trunc(+-0.5) == 0 either way.
    xx = xx + copysignf(128.0f, xx);
    return truncf(xx * (1.0f / 256.0f));
}

__global__ __launch_bounds__(256)
void armint_mlp_kernel(const float* __restrict__ x,
                       const float* __restrict__ w0, const float* __restrict__ b0,
                       const float* __restrict__ w1, const float* __restrict__ b1,
                       const float* __restrict__ wout, const float* __restrict__ bout,
                       float* __restrict__ out, int Brows) {
    __shared__ float lds[8][16 * LDSTRIDE];

    const int lane = threadIdx.x & 31;   // wave32
    const int wave = threadIdx.x >> 5;   // 8 waves per block
    const int rowBase = (blockIdx.x * 8 + wave) * 16;

    if ((int)((blockIdx.x + 1) * 128) > Brows) return;  // block-uniform guard (B divisible anyway)

    float* h = &lds[wave][0];

    // ---- Phase 0: coalesced b128 load of 16x32 tile, scale by FPFM, stage in padded LDS ----
    {
        const float4* x4 = reinterpret_cast<const float4*>(x + (size_t)rowBase * 32);
        #pragma unroll
        for (int j = 0; j < 4; ++j) {
            int idx = lane + 32 * j;          // float4 index in tile, 0..127
            float4 v = x4[idx];
            int r = idx >> 3;                 // row 0..15
            int c = (idx & 7) * 4;            // col 0,4,..,28
            float4* dst = reinterpret_cast<float4*>(&h[r * LDSTRIDE + c]);
            *dst = make_float4(v.x * FPFM, v.y * FPFM, v.z * FPFM, v.w * FPFM);
        }
    }
    __syncthreads();

    // A-fragment addressing for V_WMMA_F32_16X16X4_F32:
    //   lane L: m = L&15, holds K pair {kb, kb+1} per step, kb = 2*(L>>4)
    const int m  = lane & 15;
    const int kb = (lane >> 4) * 2;
    const int half = lane >> 4;

    const float* Ws[2] = { w0, w1 };
    const float* Bs[2] = { b0, b1 };

    #pragma unroll
    for (int layer = 0; layer < 2; ++layer) {
        const float* W    = Ws[layer];
        const float* bias = Bs[layer];

        // A fragments (shared by both N-tiles): 8 x ds_load_b64
        v2f a[8];
        #pragma unroll
        for (int s = 0; s < 8; ++s) {
            a[s] = *reinterpret_cast<const v2f*>(&h[m * LDSTRIDE + 4 * s + kb]);
        }

        float d[2][8];
        #pragma unroll
        for (int t = 0; t < 2; ++t) {
            const int ng = (lane & 15) + 16 * t;   // output column this lane covers in B/C/D layout
            v8f acc = {};
            #pragma unroll
            for (int s = 0; s < 8; ++s) {
                // B element (k, n) = W[n][k]  (h @ W.T); adjacent pair -> b64 load
                v2f b;
                b.x = W[ng * 32 + 4 * s + kb];
                b.y = W[ng * 32 + 4 * s + kb + 1];
                acc = __builtin_amdgcn_wmma_f32_16x16x4_f32(
                    /*neg_a=*/false, a[s], /*neg_b=*/false, b,
                    /*c_mod=*/(short)0, acc, /*reuse_a=*/false, /*reuse_b=*/false);
            }
            const float bn = bias[ng];
            #pragma unroll
            for (int r = 0; r < 8; ++r) {
                const int mr = r + 8 * half;       // D layout: VGPR r -> row r (lanes 0-15) / r+8
                float val = acc[r] + bn + h[mr * LDSTRIDE + ng] * FPFM;  // + residual h*FPFM
                val = round_div_fpfm(val);
                d[t][r] = fmaxf(val, 0.0f);        // ReLU
            }
        }
        __syncthreads();
        #pragma unroll
        for (int t = 0; t < 2; ++t) {
            const int ng = (lane & 15) + 16 * t;
            #pragma unroll
            for (int r = 0; r < 8; ++r) {
                const int mr = r + 8 * half;
                h[mr * LDSTRIDE + ng] = d[t][r];
            }
        }
        __syncthreads();
    }

    // ---- Output head: 2x32, one dot product per lane (row = lane&15, o = lane>>4) ----
    {
        const int row = lane & 15;
        const int o   = lane >> 4;
        float acc = bout[o];
        #pragma unroll
        for (int k = 0; k < 32; ++k)
            acc += h[row * LDSTRIDE + k] * wout[o * 32 + k];
        acc = round_div_fpfm(acc);
        const float raw = acc * (1.0f / 256.0f);
        const long long gr = (long long)rowBase + row;
        if (o == 0) {
            out[gr] = raw;                                        // mu
        } else {
            out[2LL * Brows + gr] = raw;                          // log_scale
            float e = fminf(fmaxf(raw - 4.0f, -4.6f), 5.0f);
            out[1LL * Brows + gr] = expf(e);                      // scale
        }
    }
}

extern "C" void kernel_launch(void* const* d_in, const int* in_sizes, int n_in,
                              void* d_out, int out_size, void* d_ws, size_t ws_size,
                              hipStream_t stream) {
    const float* x    = (const float*)d_in[0];
    const float* w0   = (const float*)d_in[1];
    const float* b0   = (const float*)d_in[2];
    const float* w1   = (const float*)d_in[3];
    const float* b1   = (const float*)d_in[4];
    const float* wout = (const float*)d_in[5];
    const float* bout = (const float*)d_in[6];
    float* out = (float*)d_out;

    const int Brows = in_sizes[0] / 32;      // 2097152
    const int rowsPerBlock = 128;            // 8 waves x 16 rows
    const int grid = (Brows + rowsPerBlock - 1) / rowsPerBlock;

    armint_mlp_kernel<<<grid, 256, 0, stream>>>(x, w0, b0, w1, b1, wout, bout, out, Brows);
}